// RectLoss_24275155157054
// MI455X (gfx1250) — compile-verified
//
#include <hip/hip_runtime.h>

// Problem constants (match reference)
#define BB 8
#define CC 21
#define HH 512
#define WW 512
#define HWP (HH * WW)                 // 262144 elements per (b,c) plane
#define PLANES (BB * CC)              // 168
#define BLOCKS_PER_PLANE 64           // 64 blocks * 256 thr * 16 floats = 262144
#define NPARTIAL (PLANES * BLOCKS_PER_PLANE)   // 10752
#define INTMAXV 0x7FFFFFFF

// --------------------------- CDNA5 async helpers ---------------------------
#if defined(__has_builtin)
#if __has_builtin(__builtin_amdgcn_global_load_async_to_lds_b128)
#define RL_HAVE_ASYNC_BUILTIN 1
#endif
#if __has_builtin(__builtin_amdgcn_s_wait_asynccnt)
#define RL_HAVE_WAITASYNC_BUILTIN 1
#endif
#endif

typedef int rl_v4i __attribute__((ext_vector_type(4)));
typedef __attribute__((address_space(1))) rl_v4i* rl_gp4;   // global v4i*
typedef __attribute__((address_space(3))) rl_v4i* rl_lp4;   // LDS    v4i*

__device__ __forceinline__ void rl_async_copy_b128(const float* g, float4* l) {
#if defined(RL_HAVE_ASYNC_BUILTIN)
    // signature (probe-confirmed via diagnostic):
    //   void(v4i addrspace(1)* src, v4i addrspace(3)* dst, Imm i32 offset, Imm i32 cpol)
    float* gnc = const_cast<float*>(g);
    __builtin_amdgcn_global_load_async_to_lds_b128((rl_gp4)gnc, (rl_lp4)l, 0, 0);
#else
    // GV mode: vdst = LDS byte address, vaddr = 64-bit global address, saddr = off
    unsigned ldsoff = (unsigned)(unsigned long long)l;   // flat LDS addr[31:0] = LDS offset
    asm volatile("global_load_async_to_lds_b128 %0, %1, off"
                 :: "v"(ldsoff), "v"(g)
                 : "memory");
#endif
}

template <int N>
__device__ __forceinline__ void rl_wait_async() {
#if defined(RL_HAVE_WAITASYNC_BUILTIN)
    __builtin_amdgcn_s_wait_asynccnt(N);
#else
    asm volatile("s_wait_asynccnt %0" :: "i"(N) : "memory");
#endif
    asm volatile("" ::: "memory");   // compiler barrier: no LDS read hoisting
}

// ---------------------------------------------------------------------------
// Kernel 0: init the global bbox table: {rmin=INT_MAX, rmax=-1, cmin=INT_MAX, cmax=-1}
// Layout: gbox[plane*4 + {0:rmin,1:rmax,2:cmin,3:cmax}], plane = b*CC + c.
// ---------------------------------------------------------------------------
__global__ void rl_init_kernel(int* __restrict__ gbox) {
    int i = blockIdx.x * blockDim.x + threadIdx.x;
    if (i < PLANES * 4) {
        int f = i & 3;
        gbox[i] = (f == 0 || f == 2) ? INTMAXV : -1;
    }
}

// ---------------------------------------------------------------------------
// Kernel 1: per-pixel argmax over C, then bbox (min/max row & col) per class.
// grid = (256, BB); block = 256. Each thread owns 4 consecutive pixels (float4
// per channel plane -> fully coalesced 128-bit loads). This is the HBM-bound
// pass; it also warms the 192MB L2 (tensor is 176MB) for pass 2.
// ---------------------------------------------------------------------------
__global__ __launch_bounds__(256) void rl_argmax_bbox_kernel(
        const float* __restrict__ y, int* __restrict__ gbox) {
    __shared__ int lbox[CC * 4];
    const int tid = threadIdx.x;
    if (tid < CC * 4) {
        int f = tid & 3;
        lbox[tid] = (f == 0 || f == 2) ? INTMAXV : -1;
    }
    __syncthreads();

    const int b  = blockIdx.y;
    const int p0 = (blockIdx.x * 256 + tid) * 4;     // pixel index in image
    const float* base = y + (size_t)b * CC * HWP;

    float4 best = *(const float4*)(base + p0);       // c = 0
    int i0 = 0, i1 = 0, i2 = 0, i3 = 0;
#pragma unroll
    for (int c = 1; c < CC; ++c) {
        // keep the next channel plane's line in flight (global_prefetch_b8, RT)
        if (c < CC - 1) __builtin_prefetch(base + (size_t)(c + 1) * HWP + p0, 0, 3);
        float4 v = *(const float4*)(base + (size_t)c * HWP + p0);
        if (v.x > best.x) { best.x = v.x; i0 = c; }   // strict > keeps first max
        if (v.y > best.y) { best.y = v.y; i1 = c; }
        if (v.z > best.z) { best.z = v.z; i2 = c; }
        if (v.w > best.w) { best.w = v.w; i3 = c; }
    }

    const int h  = p0 >> 9;        // W == 512
    const int w0 = p0 & 511;       // 4 pixels share the same row

    atomicMin(&lbox[i0 * 4 + 0], h);      atomicMax(&lbox[i0 * 4 + 1], h);
    atomicMin(&lbox[i0 * 4 + 2], w0);     atomicMax(&lbox[i0 * 4 + 3], w0);
    atomicMin(&lbox[i1 * 4 + 0], h);      atomicMax(&lbox[i1 * 4 + 1], h);
    atomicMin(&lbox[i1 * 4 + 2], w0 + 1); atomicMax(&lbox[i1 * 4 + 3], w0 + 1);
    atomicMin(&lbox[i2 * 4 + 0], h);      atomicMax(&lbox[i2 * 4 + 1], h);
    atomicMin(&lbox[i2 * 4 + 2], w0 + 2); atomicMax(&lbox[i2 * 4 + 3], w0 + 2);
    atomicMin(&lbox[i3 * 4 + 0], h);      atomicMax(&lbox[i3 * 4 + 1], h);
    atomicMin(&lbox[i3 * 4 + 2], w0 + 3); atomicMax(&lbox[i3 * 4 + 3], w0 + 3);

    __syncthreads();
    if (tid < CC) {
        const int P = b * CC + tid;
        atomicMin(&gbox[P * 4 + 0], lbox[tid * 4 + 0]);
        atomicMax(&gbox[P * 4 + 1], lbox[tid * 4 + 1]);
        atomicMin(&gbox[P * 4 + 2], lbox[tid * 4 + 2]);
        atomicMax(&gbox[P * 4 + 3], lbox[tid * 4 + 3]);
    }
}

// ---------------------------------------------------------------------------
// Kernel 2: BCE-with-logits against the filled bbox rectangle; deterministic
// per-block partial sums. grid = (BLOCKS_PER_PLANE, PLANES); block = 256.
// Each block streams 16KB of one (b,c) plane through LDS using CDNA5 async
// loads (GLOBAL_LOAD_ASYNC_TO_LDS_B128, ASYNCcnt-tracked, in-order done):
// all 4 stages issued up-front into distinct buffers (no WAR), drained with
// s_wait_asynccnt 3/2/1/0. Each lane reads back only its own staged float4,
// so no block barrier is needed between wait and consume.
// ---------------------------------------------------------------------------
__device__ __forceinline__ float rl_bce(float x, bool z) {
    float ax = __builtin_fabsf(x);
    float l  = fmaxf(x, 0.0f) + log1pf(__expf(-ax));
    return z ? (l - x) : l;
}

__global__ __launch_bounds__(256) void rl_loss_kernel(
        const float* __restrict__ y, const int* __restrict__ gbox,
        float* __restrict__ partials) {
    __shared__ float4 stage[4][256];                 // 16 KB of 320 KB/WGP

    const int P    = blockIdx.y;                    // b*CC + c
    const int rmin = gbox[P * 4 + 0];
    const int rmax = gbox[P * 4 + 1];
    const int cmin = gbox[P * 4 + 2];
    const int cmax = gbox[P * 4 + 3];

    const float* plane = y + (size_t)P * HWP;
    const int tid  = threadIdx.x;
    const int base = blockIdx.x * 4096;

    // Launch all 4 async stages immediately: maximum loads in flight.
#pragma unroll
    for (int k = 0; k < 4; ++k)
        rl_async_copy_b128(plane + base + (k * 256 + tid) * 4, &stage[k][tid]);

    float s = 0.0f;
#pragma unroll
    for (int it = 0; it < 4; ++it) {
        switch (it) {                                // stage `it` landed (in-order)
            case 0: rl_wait_async<3>(); break;
            case 1: rl_wait_async<2>(); break;
            case 2: rl_wait_async<1>(); break;
            default: rl_wait_async<0>(); break;
        }
        float4 v = stage[it][tid];                   // ds_load_b128, own slot
        const int e  = base + (it * 256 + tid) * 4;  // element offset in plane
        const int h  = e >> 9;
        const int w0 = e & 511;
        const bool rowin = (h >= rmin) & (h <= rmax);
        s += rl_bce(v.x, rowin & (w0     >= cmin) & (w0     <= cmax));
        s += rl_bce(v.y, rowin & (w0 + 1 >= cmin) & (w0 + 1 <= cmax));
        s += rl_bce(v.z, rowin & (w0 + 2 >= cmin) & (w0 + 2 <= cmax));
        s += rl_bce(v.w, rowin & (w0 + 3 >= cmin) & (w0 + 3 <= cmax));
    }

    // wave32 reduction (gfx1250 is wave32-only)
#pragma unroll
    for (int off = 16; off > 0; off >>= 1)
        s += __shfl_down(s, off, 32);

    __shared__ float wsum[8];
    if ((tid & 31) == 0) wsum[tid >> 5] = s;
    __syncthreads();
    if (tid == 0) {
        float t = 0.0f;
#pragma unroll
        for (int i = 0; i < 8; ++i) t += wsum[i];
        partials[P * BLOCKS_PER_PLANE + blockIdx.x] = t;
    }
}

// ---------------------------------------------------------------------------
// Kernel 3: reduce the 10752 block partials in f64 and write the mean.
// ---------------------------------------------------------------------------
__global__ __launch_bounds__(256) void rl_finalize_kernel(
        const float* __restrict__ partials, float* __restrict__ out) {
    const int tid = threadIdx.x;
    double s = 0.0;
    for (int i = tid; i < NPARTIAL; i += 256) s += (double)partials[i];
    __shared__ double sh[256];
    sh[tid] = s;
    __syncthreads();
    for (int st = 128; st > 0; st >>= 1) {
        if (tid < st) sh[tid] += sh[tid + st];
        __syncthreads();
    }
    if (tid == 0) {
        const double n = (double)BB * CC * HH * WW;   // 44040192
        out[0] = (float)(sh[0] / n);
    }
}

// ---------------------------------------------------------------------------
extern "C" void kernel_launch(void* const* d_in, const int* in_sizes, int n_in,
                              void* d_out, int out_size, void* d_ws, size_t ws_size,
                              hipStream_t stream) {
    (void)in_sizes; (void)n_in; (void)out_size; (void)ws_size;
    const float* y   = (const float*)d_in[0];
    float*       out = (float*)d_out;

    int*   gbox     = (int*)d_ws;                          // 168*4 ints (2.6 KB)
    float* partials = (float*)((char*)d_ws + 4096);        // 10752 floats (42 KB)

    rl_init_kernel<<<dim3(3), dim3(256), 0, stream>>>(gbox);
    rl_argmax_bbox_kernel<<<dim3(256, BB), dim3(256), 0, stream>>>(y, gbox);
    rl_loss_kernel<<<dim3(BLOCKS_PER_PLANE, PLANES), dim3(256), 0, stream>>>(y, gbox, partials);
    rl_finalize_kernel<<<dim3(1), dim3(256), 0, stream>>>(partials, out);
}